// TuckEREmbedding_3753801416764
// MI455X (gfx1250) — compile-verified
//
#include <hip/hip_runtime.h>

typedef __attribute__((ext_vector_type(2))) float v2f;
typedef __attribute__((ext_vector_type(8))) float v8f;

#define B_DIM 64
#define H_DIM 768
#define E_DIM 128

// ---------------------------------------------------------------------------
// Kernel 1: t[i] = sum_{j,k} core[i][j][k]   (streams the dominant 8 MB once)
// 128 blocks (one per i) x 256 threads, float4 (global_load_b128) loads.
// ---------------------------------------------------------------------------
__global__ __launch_bounds__(256) void core_reduce_kernel(
    const float* __restrict__ core, float* __restrict__ t) {
  const int i = blockIdx.x;
  const float4* p =
      reinterpret_cast<const float4*>(core + (size_t)i * (E_DIM * E_DIM));
  float s = 0.f;
#pragma unroll 4
  for (int idx = threadIdx.x; idx < (E_DIM * E_DIM) / 4; idx += 256) {
    float4 v = p[idx];
    s += (v.x + v.y) + (v.z + v.w);
  }
  __shared__ float red[256];
  red[threadIdx.x] = s;
  __syncthreads();
  for (int off = 128; off > 0; off >>= 1) {
    if (threadIdx.x < off) red[threadIdx.x] += red[threadIdx.x + off];
    __syncthreads();
  }
  if (threadIdx.x == 0) t[i] = red[0];
}

// ---------------------------------------------------------------------------
// Kernel 2: P = S @ W^T + bias via V_WMMA_F32_16X16X4_F32 (exact fp32 path).
// One wave per 16x16 output tile. blockIdx.z selects projection:
//   z=0: head = head_src @ W_e^T + b_e
//   z=1: tail = tail_src @ W_e^T + b_e
//   z=2: rel  = rel_src  @ W_r^T + b_r
//
// ISA fragment layouts (32-bit, wave32):
//   A 16x4 : lanes 0-15 -> M=lane, v0/v1 = K=0/1 ; lanes 16-31 -> K=2/3
//   B 4x16 : lanes 0-15 -> N=lane, v0/v1 = K=0/1 ; lanes 16-31 -> K=2/3
//   C/D    : VGPR r, lanes 0-15: (M=r, N=lane); lanes 16-31: (M=r+8, N=lane-16)
// Both A and B fragments become the same per-lane 8B-aligned float2 load from
// row-major S / W (W row n is column n of W^T).
// ---------------------------------------------------------------------------
__global__ __launch_bounds__(32) void proj_wmma_kernel(
    const float* __restrict__ s0, const float* __restrict__ s1,
    const float* __restrict__ s2, const float* __restrict__ We,
    const float* __restrict__ be, const float* __restrict__ Wr,
    const float* __restrict__ br, float* __restrict__ out /* [3][64][128] */) {
  const int z = blockIdx.z;
  const float* S = (z == 0) ? s0 : (z == 1) ? s1 : s2;
  const float* W = (z == 2) ? Wr : We;
  const float* bias = (z == 2) ? br : be;
  float* P = out + (size_t)z * (B_DIM * E_DIM);

  const int lane = threadIdx.x;  // 0..31
  const int half = lane >> 4;    // 0: K=0,1  1: K=2,3
  const int lr = lane & 15;
  const int m0 = blockIdx.x * 16;
  const int n0 = blockIdx.y * 16;

  const float* arow = S + (size_t)(m0 + lr) * H_DIM + 2 * half;
  const float* brow = W + (size_t)(n0 + lr) * H_DIM + 2 * half;

  v8f acc = {};
#pragma unroll 8
  for (int k = 0; k < H_DIM; k += 4) {
    v2f a = *reinterpret_cast<const v2f*>(arow + k);
    v2f b = *reinterpret_cast<const v2f*>(brow + k);
    // 8 args: (neg_a, A, neg_b, B, c_mod, C, reuse_a, reuse_b)
    acc = __builtin_amdgcn_wmma_f32_16x16x4_f32(false, a, false, b, (short)0,
                                                acc, false, false);
  }

  const float bv = bias[n0 + lr];
#pragma unroll
  for (int r = 0; r < 8; ++r) {
    const int m = m0 + r + 8 * half;
    P[(size_t)m * E_DIM + n0 + lr] = acc[r] + bv;
  }
}

// ---------------------------------------------------------------------------
// Kernel 3: energy[b] = -sum_i head[b,i]*rel[b,i]*tail[b,i]*t[i]
// ---------------------------------------------------------------------------
__global__ __launch_bounds__(128) void energy_kernel(
    const float* __restrict__ proj, const float* __restrict__ t,
    float* __restrict__ out) {
  const int b = blockIdx.x;
  const int i = threadIdx.x;  // 0..127
  const float* head = proj;
  const float* tail = proj + B_DIM * E_DIM;
  const float* rel = proj + 2 * B_DIM * E_DIM;
  const size_t o = (size_t)b * E_DIM + i;
  float v = head[o] * rel[o] * tail[o] * t[i];
  __shared__ float red[128];
  red[i] = v;
  __syncthreads();
  for (int off = 64; off > 0; off >>= 1) {
    if (i < off) red[i] += red[i + off];
    __syncthreads();
  }
  if (i == 0) out[b] = -red[0];
}

extern "C" void kernel_launch(void* const* d_in, const int* in_sizes, int n_in,
                              void* d_out, int out_size, void* d_ws,
                              size_t ws_size, hipStream_t stream) {
  const float* head_src = (const float*)d_in[0];
  const float* rel_src = (const float*)d_in[1];
  const float* tail_src = (const float*)d_in[2];
  const float* W_e = (const float*)d_in[3];
  const float* b_e = (const float*)d_in[4];
  const float* W_r = (const float*)d_in[5];
  const float* b_r = (const float*)d_in[6];
  const float* core = (const float*)d_in[7];

  float* t = (float*)d_ws;    // 128 floats
  float* proj = t + E_DIM;    // 3 * 64 * 128 floats (head, tail, rel)

  // 1) reduce core -> t  (bandwidth-dominant: 8 MB single pass)
  core_reduce_kernel<<<dim3(E_DIM), dim3(256), 0, stream>>>(core, t);

  // 2) three fp32 WMMA projections: grid (Mtiles=4, Ntiles=8, proj=3)
  proj_wmma_kernel<<<dim3(B_DIM / 16, E_DIM / 16, 3), dim3(32), 0, stream>>>(
      head_src, tail_src, rel_src, W_e, b_e, W_r, b_r, proj);

  // 3) fold into energy[b]
  energy_kernel<<<dim3(B_DIM), dim3(128), 0, stream>>>(proj, t, (float*)d_out);
}